// callPredictor_26396869001740
// MI455X (gfx1250) — compile-verified
//
#include <hip/hip_runtime.h>
#include <hip/hip_bf16.h>
#include <math.h>

// ---------------------------------------------------------------------------
// RETAIN predictor for MI455X (gfx1250), fp32 end-to-end using
// V_WMMA_F32_16X16X4_F32 for all dense matmuls. Attention uses the
// suffix-sum identity (flag = triu => reverse cumsum) to avoid the [B,S,S]
// matmul entirely.
// B=16, S=256, F=512, E=128, H=64, 3H=192, L=8
// ---------------------------------------------------------------------------

typedef __attribute__((ext_vector_type(2))) float v2f;
typedef __attribute__((ext_vector_type(8))) float v8f;

#define BATCH 16
#define SEQ   256
#define FEAT  512
#define EMB   128
#define HID   64
#define G3    192
#define LBL   8

// ---------------------------------------------------------------------------
// Generic fp32 WMMA GEMM:  D[M,N] = A[M,K] * W[N,K]^T + bias[N]   (opt tanh)
// One wave computes one 16x16 tile; 4 waves per block. M,N multiples of 16,
// K multiple of 4. Early-exit is wave-uniform so EXEC stays all-1 for WMMA.
// ---------------------------------------------------------------------------
__global__ __launch_bounds__(128) void gemm_nt_f32(
    const float* __restrict__ A, int lda,
    const float* __restrict__ W, int ldw,
    const float* __restrict__ bias,
    float* __restrict__ D, int ldd,
    int M, int N, int K, int act)
{
    const int wave = threadIdx.x >> 5;
    const int lane = threadIdx.x & 31;
    const int tilesN = N >> 4;
    const int numTiles = (M >> 4) * tilesN;
    const int tileId = blockIdx.x * 4 + wave;
    if (tileId >= numTiles) return;                 // uniform per wave

    const int m0 = (tileId / tilesN) << 4;
    const int n0 = (tileId % tilesN) << 4;

    const int row = lane & 15;                      // M index (A) / N index (B)
    const int kb  = (lane >> 4) << 1;               // 0 or 2 (K sub-offset)

    const float* Arow = A + (size_t)(m0 + row) * lda + kb;
    const float* Wrow = W + (size_t)(n0 + row) * ldw + kb;

    v8f c = {};
    for (int k0 = 0; k0 < K; k0 += 4) {
        v2f a, b;
        a.x = Arow[k0];
        a.y = Arow[k0 + 1];
        b.x = Wrow[k0];
        b.y = Wrow[k0 + 1];
        c = __builtin_amdgcn_wmma_f32_16x16x4_f32(false, a, false, b,
                                                  (short)0, c, false, false);
    }

    const int col   = lane & 15;
    const int mbase = m0 + ((lane >> 4) << 3);
    const float bv  = bias ? bias[n0 + col] : 0.0f;
    #pragma unroll
    for (int v = 0; v < 8; ++v) {
        float val = c[v] + bv;
        if (act == 1) val = tanhf(val);
        D[(size_t)(mbase + v) * ldd + n0 + col] = val;
    }
}

// ---------------------------------------------------------------------------
// GRU scan. One block per GRU (blockIdx.x: 0 = alpha, 1 = beta), 12 waves.
// Per step: gh[16,192] = h[16,64] @ Whh^T via 12 waves x 16 WMMA(16x16x4).
// Whh fragments live in registers for the whole scan; h lives in LDS.
// Waves 0-3 -> r gate, 4-7 -> z gate, 8-11 -> n gate + h update.
// ---------------------------------------------------------------------------
__global__ __launch_bounds__(384) void gru_scan(
    const float* __restrict__ xp_a, const float* __restrict__ xp_b,
    const float* __restrict__ Whh_a, const float* __restrict__ Whh_b,
    const float* __restrict__ bhh_a, const float* __restrict__ bhh_b,
    float* __restrict__ hout_a, float* __restrict__ hout_b)
{
    const float* xp   = (blockIdx.x == 0) ? xp_a  : xp_b;
    const float* Whh  = (blockIdx.x == 0) ? Whh_a : Whh_b;
    const float* bhh  = (blockIdx.x == 0) ? bhh_a : bhh_b;
    float*       hout = (blockIdx.x == 0) ? hout_a : hout_b;

    __shared__ float h_s[BATCH * HID];   // current hidden state
    __shared__ float r_s[BATCH * HID];   // reset gate
    __shared__ float z_s[BATCH * HID];   // update gate

    const int wave = threadIdx.x >> 5;   // 0..11 -> N-tile (g range)
    const int lane = threadIdx.x & 31;
    const int n0   = wave << 4;
    const int col  = lane & 15;
    const int row  = lane & 15;
    const int kb   = (lane >> 4) << 1;
    const int mb   = (lane >> 4) << 3;   // batch-row base for C/D frags

    // Preload B fragments: Bmat[k][n] = Whh[n0+col][k], 16 K-steps of 4.
    v2f bw[16];
    #pragma unroll
    for (int ks = 0; ks < 16; ++ks) {
        const float* wrow = Whh + (size_t)(n0 + col) * HID + ks * 4 + kb;
        bw[ks].x = wrow[0];
        bw[ks].y = wrow[1];
    }
    const float bhv = bhh[n0 + col];

    for (int i = threadIdx.x; i < BATCH * HID; i += blockDim.x) h_s[i] = 0.0f;
    __syncthreads();

    for (int t = 0; t < SEQ; ++t) {
        // gh tile = h @ Whh^T  (16 chained fp32 WMMAs)
        v8f c = {};
        #pragma unroll
        for (int ks = 0; ks < 16; ++ks) {
            v2f a;
            a.x = h_s[row * HID + ks * 4 + kb];
            a.y = h_s[row * HID + ks * 4 + kb + 1];
            c = __builtin_amdgcn_wmma_f32_16x16x4_f32(false, a, false, bw[ks],
                                                      (short)0, c, false, false);
        }

        if (wave < 8) {
            // r = sigmoid(xr + hr)   /   z = sigmoid(xz + hz)
            float* dst   = (wave < 4) ? r_s : z_s;
            const int hidx = (wave < 4) ? (n0 + col) : (n0 + col - HID);
            #pragma unroll
            for (int v = 0; v < 8; ++v) {
                const int b = mb + v;
                const float x = xp[((size_t)b * SEQ + t) * G3 + n0 + col];
                const float g = c[v] + bhv + x;
                dst[b * HID + hidx] = 1.0f / (1.0f + __expf(-g));
            }
            __syncthreads();   // A: r,z published; all h reads of this step done
            __syncthreads();   // B: h updated by waves 8-11
        } else {
            // n = tanh(xn + r*hn);  h' = (1-z)*n + z*h
            const int hidx = n0 + col - 2 * HID;
            float hn[8], xn[8];
            #pragma unroll
            for (int v = 0; v < 8; ++v) {
                const int b = mb + v;
                hn[v] = c[v] + bhv;
                xn[v] = xp[((size_t)b * SEQ + t) * G3 + n0 + col];
            }
            __syncthreads();   // A
            #pragma unroll
            for (int v = 0; v < 8; ++v) {
                const int b  = mb + v;
                const float r  = r_s[b * HID + hidx];
                const float z  = z_s[b * HID + hidx];
                const float ho = h_s[b * HID + hidx];
                const float n  = tanhf(xn[v] + r * hn[v]);
                const float hv = (1.0f - z) * n + z * ho;
                h_s[b * HID + hidx] = hv;
                hout[((size_t)b * SEQ + t) * HID + hidx] = hv;
            }
            __syncthreads();   // B
        }
    }
}

// ---------------------------------------------------------------------------
// attw[b,s] = exp(h_alpha[b,s,:].Wa + ba) * M[b,s]
// ---------------------------------------------------------------------------
__global__ void corr_kernel(const float* __restrict__ h_a,
                            const float* __restrict__ Wa,
                            const float* __restrict__ ba,
                            const float* __restrict__ M,
                            float* __restrict__ attw)
{
    const int idx = blockIdx.x * blockDim.x + threadIdx.x;   // b*S+s
    if (idx >= BATCH * SEQ) return;
    const float* h = h_a + (size_t)idx * HID;
    float acc = ba[0];
    #pragma unroll 8
    for (int k = 0; k < HID; ++k) acc += h[k] * Wa[k];
    attw[idx] = __expf(acc) * M[idx];
}

// ---------------------------------------------------------------------------
// Attention via suffix scan:  weighted[b,i,:] =
//   (sum_{j>=i} attw[b,j]*beta_w[b,j,:]*emb[b,j,:]) / (sum_{j>=i} attw[b,j]+eps)
// One block per batch, one thread per embedding channel.
// ---------------------------------------------------------------------------
__global__ __launch_bounds__(128) void attn_weighted(
    const float* __restrict__ attw,
    const float* __restrict__ betaw,
    const float* __restrict__ emb,
    float* __restrict__ weighted)
{
    const int b = blockIdx.x;
    const int e = threadIdx.x;   // 0..127
    float cumv = 0.0f, cumw = 0.0f;
    for (int j = SEQ - 1; j >= 0; --j) {
        const float w = attw[b * SEQ + j];
        const size_t idx = ((size_t)b * SEQ + j) * EMB + e;
        cumv += w * betaw[idx] * emb[idx];
        cumw += w;
        weighted[idx] = cumv / (cumw + 1e-10f);
    }
}

// ---------------------------------------------------------------------------
// all_output[b,s,l] = (weighted[b,s,:].Wp[l,:] + bp[l]) * M[b,s]
// ---------------------------------------------------------------------------
__global__ void predict_kernel(const float* __restrict__ weighted,
                               const float* __restrict__ Wp,
                               const float* __restrict__ bp,
                               const float* __restrict__ M,
                               float* __restrict__ out)
{
    const int idx = blockIdx.x * blockDim.x + threadIdx.x;   // (b*S+s)*L + l
    if (idx >= BATCH * SEQ * LBL) return;
    const int l  = idx & (LBL - 1);
    const int bs = idx >> 3;
    const float* wr = weighted + (size_t)bs * EMB;
    const float* wp = Wp + (size_t)l * EMB;
    float acc = bp[l];
    #pragma unroll 8
    for (int k = 0; k < EMB; ++k) acc += wr[k] * wp[k];
    out[idx] = acc * M[bs];
}

// ---------------------------------------------------------------------------
// cur_output[b,l] = sum_s all_output[b,s,l] * cur_M[b,s]
// ---------------------------------------------------------------------------
__global__ __launch_bounds__(128) void cur_kernel(const float* __restrict__ all_out,
                                                  const float* __restrict__ cur_M,
                                                  float* __restrict__ out)
{
    const int idx = threadIdx.x;          // b*L + l, 0..127
    const int b = idx >> 3, l = idx & 7;
    float acc = 0.0f;
    for (int s = 0; s < SEQ; ++s)
        acc += all_out[((size_t)b * SEQ + s) * LBL + l] * cur_M[b * SEQ + s];
    out[(size_t)BATCH * SEQ * LBL + idx] = acc;
}

// ---------------------------------------------------------------------------
extern "C" void kernel_launch(void* const* d_in, const int* in_sizes, int n_in,
                              void* d_out, int out_size, void* d_ws, size_t ws_size,
                              hipStream_t stream)
{
    const float* X       = (const float*)d_in[0];
    const float* M       = (const float*)d_in[1];
    const float* cur_M   = (const float*)d_in[2];
    const float* W_embed = (const float*)d_in[3];
    const float* b_embed = (const float*)d_in[4];
    const float* Wih_a   = (const float*)d_in[5];
    const float* Whh_a   = (const float*)d_in[6];
    const float* bih_a   = (const float*)d_in[7];
    const float* bhh_a   = (const float*)d_in[8];
    const float* Wih_b   = (const float*)d_in[9];
    const float* Whh_b   = (const float*)d_in[10];
    const float* bih_b   = (const float*)d_in[11];
    const float* bhh_b   = (const float*)d_in[12];
    const float* Wb      = (const float*)d_in[13];
    const float* bb      = (const float*)d_in[14];
    const float* Wa      = (const float*)d_in[15];
    const float* ba      = (const float*)d_in[16];
    const float* Wp      = (const float*)d_in[17];
    const float* bp      = (const float*)d_in[18];

    float* out = (float*)d_out;

    // Workspace layout (floats)
    float* ws       = (float*)d_ws;
    float* emb      = ws;                               // 4096*128
    float* xp_a     = emb      + (size_t)4096 * 128;    // 4096*192
    float* xp_b     = xp_a     + (size_t)4096 * 192;    // 4096*192
    float* h_a      = xp_b     + (size_t)4096 * 192;    // 4096*64
    float* h_b      = h_a      + (size_t)4096 * 64;     // 4096*64
    float* betaw    = h_b      + (size_t)4096 * 64;     // 4096*128
    float* attw     = betaw    + (size_t)4096 * 128;    // 4096
    float* weighted = attw     + (size_t)4096;          // 4096*128

    const int MROWS = BATCH * SEQ;   // 4096

    // 1) emb = X @ W_embed^T + b_embed        [4096,128] = [4096,512]x[128,512]^T
    {
        int tiles = (MROWS / 16) * (EMB / 16);
        gemm_nt_f32<<<(tiles + 3) / 4, 128, 0, stream>>>(
            X, FEAT, W_embed, FEAT, b_embed, emb, EMB, MROWS, EMB, FEAT, 0);
    }
    // 2) xp_a / xp_b = emb @ Wih^T + bih      [4096,192]
    {
        int tiles = (MROWS / 16) * (G3 / 16);
        gemm_nt_f32<<<(tiles + 3) / 4, 128, 0, stream>>>(
            emb, EMB, Wih_a, EMB, bih_a, xp_a, G3, MROWS, G3, EMB, 0);
        gemm_nt_f32<<<(tiles + 3) / 4, 128, 0, stream>>>(
            emb, EMB, Wih_b, EMB, bih_b, xp_b, G3, MROWS, G3, EMB, 0);
    }
    // 3) Both GRU scans concurrently (one block each)
    gru_scan<<<2, 384, 0, stream>>>(xp_a, xp_b, Whh_a, Whh_b, bhh_a, bhh_b, h_a, h_b);

    // 4) beta_w = tanh(h_beta @ Wb^T + bb)    [4096,128], K=64
    {
        int tiles = (MROWS / 16) * (EMB / 16);
        gemm_nt_f32<<<(tiles + 3) / 4, 128, 0, stream>>>(
            h_b, HID, Wb, HID, bb, betaw, EMB, MROWS, EMB, HID, 1);
    }
    // 5) attw = exp(h_alpha . Wa + ba) * M
    corr_kernel<<<(MROWS + 255) / 256, 256, 0, stream>>>(h_a, Wa, ba, M, attw);

    // 6) suffix-scan attention -> weighted    [4096,128]
    attn_weighted<<<BATCH, 128, 0, stream>>>(attw, betaw, emb, weighted);

    // 7) all_output = (weighted @ Wp^T + bp) * M  -> d_out[0 .. 32767]
    predict_kernel<<<(MROWS * LBL + 255) / 256, 256, 0, stream>>>(
        weighted, Wp, bp, M, out);

    // 8) cur_output -> d_out[32768 .. 32895]
    cur_kernel<<<1, 128, 0, stream>>>(out, cur_M, out);
}